// RNN_51780125721510
// MI455X (gfx1250) — compile-verified
//
#include <hip/hip_runtime.h>
#include <hip/hip_bf16.h>

#define N 4096
#define LN_EPS 1e-5f
#define NSLICE 32              // K-split for the column GEMVs
#define KS (N / NSLICE)        // 128 K per slice

typedef __attribute__((ext_vector_type(2))) float v2f;
typedef __attribute__((ext_vector_type(4))) float v4f;
typedef __attribute__((ext_vector_type(8))) float v8f;

// ---------------------------------------------------------------------------
// K1: wx[j] = relu( sum_i x[i] * w_in[j,i] )   (row-major dot, 1 wave / row)
// ---------------------------------------------------------------------------
__global__ void k_wx(const float* __restrict__ x, const float* __restrict__ w_in,
                     float* __restrict__ wx) {
  const int lane = threadIdx.x & 31;
  const int row  = (int)((blockIdx.x * blockDim.x + threadIdx.x) >> 5);
  const v4f* __restrict__ wr = (const v4f*)(w_in + (size_t)row * N);
  const v4f* __restrict__ xv = (const v4f*)x;
  float acc = 0.f;
#pragma unroll 4
  for (int t = lane; t < N / 4; t += 32) {
    v4f a = xv[t];
    v4f b = __builtin_nontemporal_load(wr + t);   // streamed once: NT
    acc += a.x * b.x + a.y * b.y + a.z * b.z + a.w * b.w;
  }
#pragma unroll
  for (int off = 16; off > 0; off >>= 1) acc += __shfl_xor(acc, off, 32);
  if (lane == 0) wx[row] = fmaxf(acc, 0.f);
}

// ---------------------------------------------------------------------------
// K2: partial GEMV wh = hid @ (w + alpha*hebb) via V_WMMA_F32_16X16X4_F32.
// One wave handles 16 output columns x a 128-wide K slice.
//   A (16x4, f32): A[m,k] = hid[kbase+k] for every m  (rows replicated)
//   B (4x16, f32): B[k,n] = w[.]+alpha[.]*hebb[.]  at row kbase+k, col n0+n
//   D row 0 = partial sums for the 16 columns.
// ---------------------------------------------------------------------------
__global__ void k_wh_part(const float* __restrict__ hid, const float* __restrict__ w,
                          const float* __restrict__ alpha, const float* __restrict__ hebb,
                          float* __restrict__ part) {
  const int lane  = threadIdx.x & 31;
  const int wid   = (int)((blockIdx.x * blockDim.x + threadIdx.x) >> 5);
  const int cg    = wid >> 5;          // column group 0..255
  const int slice = wid & (NSLICE - 1);
  const int nc    = cg * 16 + (lane & 15);
  const int khalf = (lane < 16) ? 0 : 2;
  const int kbase = slice * KS;
  v8f acc = {};
  for (int k0 = kbase; k0 < kbase + KS; k0 += 4) {
    const int kr = k0 + khalf;
    v2f A;
    A.x = hid[kr];
    A.y = hid[kr + 1];
    const size_t o0 = (size_t)kr * N + nc;
    const size_t o1 = o0 + N;
    v2f B;                                             // w,alpha: NT (read once)
    B.x = fmaf(__builtin_nontemporal_load(alpha + o0), hebb[o0],
               __builtin_nontemporal_load(w + o0));    // hebb: RT, reused by K5
    B.y = fmaf(__builtin_nontemporal_load(alpha + o1), hebb[o1],
               __builtin_nontemporal_load(w + o1));
    acc = __builtin_amdgcn_wmma_f32_16x16x4_f32(false, A, false, B,
                                                (short)0, acc, false, false);
  }
  if (lane < 16) part[(size_t)slice * N + nc] = acc[0];   // D row 0
}

// ---------------------------------------------------------------------------
// K3: partial GEMV z = wh @ pred_eta  (same WMMA scheme)
// ---------------------------------------------------------------------------
__global__ void k_eta_part(const float* __restrict__ wh, const float* __restrict__ pe,
                           float* __restrict__ part) {
  const int lane  = threadIdx.x & 31;
  const int wid   = (int)((blockIdx.x * blockDim.x + threadIdx.x) >> 5);
  const int cg    = wid >> 5;
  const int slice = wid & (NSLICE - 1);
  const int nc    = cg * 16 + (lane & 15);
  const int khalf = (lane < 16) ? 0 : 2;
  const int kbase = slice * KS;
  v8f acc = {};
  for (int k0 = kbase; k0 < kbase + KS; k0 += 4) {
    const int kr = k0 + khalf;
    v2f A;
    A.x = wh[kr];
    A.y = wh[kr + 1];
    const size_t o0 = (size_t)kr * N + nc;
    v2f B;
    B.x = __builtin_nontemporal_load(pe + o0);
    B.y = __builtin_nontemporal_load(pe + o0 + N);
    acc = __builtin_amdgcn_wmma_f32_16x16x4_f32(false, A, false, B,
                                                (short)0, acc, false, false);
  }
  if (lane < 16) part[(size_t)slice * N + nc] = acc[0];
}

// ---------------------------------------------------------------------------
// Reduce K-slices: out[j] = bias[j] + sum_s part[s][j]; mode 1 applies sigmoid
// ---------------------------------------------------------------------------
__global__ void k_reduce(const float* __restrict__ part, const float* __restrict__ bias,
                         float* __restrict__ out, int mode) {
  const int j = (int)(blockIdx.x * blockDim.x + threadIdx.x);
  float s = bias[j];
#pragma unroll
  for (int sl = 0; sl < NSLICE; ++sl) s += part[(size_t)sl * N + j];
  if (mode == 1) s = 1.f / (1.f + __expf(-s));
  out[j] = s;
}

// ---------------------------------------------------------------------------
// K4: s = wx + wh ; LayerNorm ; ReLU  (single 1024-thread block, 4 elem/thr)
// ---------------------------------------------------------------------------
__global__ void k_ln(const float* __restrict__ wx, const float* __restrict__ wh,
                     const float* __restrict__ gamma, const float* __restrict__ beta,
                     float* __restrict__ wy) {
  __shared__ float red[64];
  const int base = threadIdx.x * 4;
  v4f a = *(const v4f*)(wx + base);
  v4f b = *(const v4f*)(wh + base);
  v4f s = a + b;
  float sum = s.x + s.y + s.z + s.w;
  float sq  = s.x * s.x + s.y * s.y + s.z * s.z + s.w * s.w;
#pragma unroll
  for (int off = 16; off > 0; off >>= 1) {
    sum += __shfl_xor(sum, off, 32);
    sq  += __shfl_xor(sq,  off, 32);
  }
  const int wid = threadIdx.x >> 5, lane = threadIdx.x & 31;
  if (lane == 0) { red[wid] = sum; red[32 + wid] = sq; }
  __syncthreads();
  if (wid == 0) {
    float t0 = red[lane], t1 = red[32 + lane];
#pragma unroll
    for (int off = 16; off > 0; off >>= 1) {
      t0 += __shfl_xor(t0, off, 32);
      t1 += __shfl_xor(t1, off, 32);
    }
    if (lane == 0) { red[0] = t0; red[32] = t1; }
  }
  __syncthreads();
  const float mu  = red[0] * (1.f / N);
  const float var = red[32] * (1.f / N) - mu * mu;
  const float inv = rsqrtf(var + LN_EPS);
  v4f g = *(const v4f*)(gamma + base);
  v4f e = *(const v4f*)(beta + base);
  v4f r;
  r.x = fmaxf(g.x * (s.x - mu) * inv + e.x, 0.f);
  r.y = fmaxf(g.y * (s.y - mu) * inv + e.y, 0.f);
  r.z = fmaxf(g.z * (s.z - mu) * inv + e.z, 0.f);
  r.w = fmaxf(g.w * (s.w - mu) * inv + e.w, 0.f);
  *(v4f*)(wy + base) = r;
}

// ---------------------------------------------------------------------------
// K5: hebb'[i,j] = (1-lam)*hebb[i,j] + (eta[i]*hid[i]) * wh[j]
// ---------------------------------------------------------------------------
__global__ void k_hebb(const float* __restrict__ hebb, const float* __restrict__ hid,
                       const float* __restrict__ wh, const float* __restrict__ eta,
                       const float* __restrict__ lamda, float* __restrict__ out) {
  const size_t tid = (size_t)blockIdx.x * blockDim.x + threadIdx.x;
  const int i = (int)(tid >> 10);            // 1024 threads per row (4 cols each)
  const int j = (int)(tid & 1023) << 2;
  const float keep = 1.f - lamda[0];
  const float c = eta[i] * hid[i];
  const size_t off = ((size_t)i << 12) + (size_t)j;
  v4f h  = *(const v4f*)(hebb + off);        // RT: may still sit in 192MB L2
  v4f wv = *(const v4f*)(wh + j);
  v4f o  = keep * h + c * wv;
  __builtin_nontemporal_store(o, (v4f*)(out + off));   // never re-read
}

// ---------------------------------------------------------------------------
extern "C" void kernel_launch(void* const* d_in, const int* in_sizes, int n_in,
                              void* d_out, int out_size, void* d_ws, size_t ws_size,
                              hipStream_t stream) {
  const float* x          = (const float*)d_in[0];
  const float* hid        = (const float*)d_in[1];
  const float* hebb       = (const float*)d_in[2];
  const float* w_in       = (const float*)d_in[3];
  const float* w          = (const float*)d_in[4];
  const float* alpha      = (const float*)d_in[5];
  const float* b          = (const float*)d_in[6];
  const float* lamda      = (const float*)d_in[7];
  const float* pred_eta   = (const float*)d_in[8];
  const float* pred_eta_b = (const float*)d_in[9];
  const float* ln_gamma   = (const float*)d_in[10];
  const float* ln_beta    = (const float*)d_in[11];

  float* out      = (float*)d_out;
  float* wy       = out;           // [4096]
  float* hebb_new = out + N;       // [4096*4096]

  float* ws   = (float*)d_ws;
  float* wx   = ws;                // [N]
  float* wh   = ws + N;            // [N]
  float* eta  = ws + 2 * N;        // [N]
  float* part = ws + 3 * N;        // [NSLICE * N] partial sums (512 KB)

  // wx (independent of wh chain)
  k_wx<<<N / 8, 256, 0, stream>>>(x, w_in, wx);                 // 4096 waves

  // wh = hid @ (w + alpha*hebb) + b     (8192 waves over 192MB stream)
  k_wh_part<<<(256 * NSLICE) / 8, 256, 0, stream>>>(hid, w, alpha, hebb, part);
  k_reduce<<<N / 256, 256, 0, stream>>>(part, b, wh, 0);

  // eta = sigmoid(wh @ pred_eta + pred_eta_b)
  k_eta_part<<<(256 * NSLICE) / 8, 256, 0, stream>>>(wh, pred_eta, part);
  k_reduce<<<N / 256, 256, 0, stream>>>(part, pred_eta_b, eta, 1);

  // wy = relu(LayerNorm(wx + wh))
  k_ln<<<1, 1024, 0, stream>>>(wx, wh, ln_gamma, ln_beta, wy);

  // hebb update (16M elements, float4 per thread)
  k_hebb<<<(N * (N / 4)) / 256, 256, 0, stream>>>(hebb, hid, wh, eta, lamda, hebb_new);
}